// StatelessMambaDecoder_84628035600838
// MI455X (gfx1250) — compile-verified
//
#include <hip/hip_runtime.h>
#include <hip/hip_bf16.h>

typedef float v2f __attribute__((ext_vector_type(2)));
typedef float v8f __attribute__((ext_vector_type(8)));

#define NBATCH 8
#define NSEQ   4096
#define DM     256
#define DI     512          // 2*DM
#define NST    16           // D_STATE
#define DS2    33           // 2*NST + 1
#define MROWS  (NBATCH * NSEQ)   // 32768
#define LEPS   1e-5f

__device__ __forceinline__ float wave_sum32(float v) {
    v += __shfl_xor(v, 16);
    v += __shfl_xor(v, 8);
    v += __shfl_xor(v, 4);
    v += __shfl_xor(v, 2);
    v += __shfl_xor(v, 1);
    return v;
}

__device__ __forceinline__ float silu_f(float x) {
    return x / (1.0f + __expf(-x));
}

// ---------------------------------------------------------------------------
// x[m,c] = sum_j z[m,j]*ip_w[c,j] + ip_b[c]   (K=4)
// ---------------------------------------------------------------------------
__global__ void k_input_proj(const float* __restrict__ z, const float* __restrict__ w,
                             const float* __restrict__ b, float* __restrict__ x) {
    int idx = blockIdx.x * blockDim.x + threadIdx.x;       // MROWS*DM threads
    int m = idx >> 8;
    int c = idx & 255;
    float4 zv = *(const float4*)(z + (size_t)m * 4);
    float4 wv = *(const float4*)(w + (size_t)c * 4);
    x[(size_t)m * DM + c] = zv.x * wv.x + zv.y * wv.y + zv.z * wv.z + zv.w * wv.w + b[c];
}

// ---------------------------------------------------------------------------
// Wave-per-row layernorm over 256 columns
// ---------------------------------------------------------------------------
__global__ void k_layernorm(const float* __restrict__ x, const float* __restrict__ g,
                            const float* __restrict__ b, float* __restrict__ o, int M) {
    int wave = (blockIdx.x * blockDim.x + threadIdx.x) >> 5;
    int lane = threadIdx.x & 31;
    if (wave >= M) return;
    const float* row = x + (size_t)wave * DM;
    float v[8];
    float s = 0.0f;
#pragma unroll
    for (int i = 0; i < 8; ++i) { v[i] = row[i * 32 + lane]; s += v[i]; }
    s = wave_sum32(s);
    float mu = s * (1.0f / DM);
    float q = 0.0f;
#pragma unroll
    for (int i = 0; i < 8; ++i) { float d = v[i] - mu; q += d * d; }
    q = wave_sum32(q);
    float rs = rsqrtf(q * (1.0f / DM) + LEPS);
    float* orow = o + (size_t)wave * DM;
#pragma unroll
    for (int i = 0; i < 8; ++i) {
        int c = i * 32 + lane;
        orow[c] = (v[i] - mu) * rs * g[c] + b[c];
    }
}

// ---------------------------------------------------------------------------
// WMMA f32 GEMM:  O[m,n] = sum_k A[m,k] * W[n,k]  (+ R[m,n] if R != nullptr)
// A: MxK row-major, W: NxK row-major. Wave owns a 16x64 output strip
// (A fragment reused across 4 N-tiles). N must be a multiple of 64.
// ---------------------------------------------------------------------------
__global__ void k_wmma_gemm_nt(const float* __restrict__ A, const float* __restrict__ W,
                               const float* __restrict__ R, float* __restrict__ O,
                               int M, int N, int K) {
    int wavesPerBlock = blockDim.x >> 5;
    int waveId = blockIdx.x * wavesPerBlock + (threadIdx.x >> 5);
    int tilesN4 = N >> 6;                       // groups of 4 N-tiles
    int totalWaves = (M >> 4) * tilesN4;
    if (waveId >= totalWaves) return;
    int tm  = waveId / tilesN4;
    int tn4 = waveId % tilesN4;
    int lane = threadIdx.x & 31;
    int half = lane >> 4;                       // 0 or 1 (selects K pair / row group)
    int r    = lane & 15;

    // A fragment source: A[tm*16 + r][k + 2*half + {0,1}]
    const float* arow = A + (size_t)(tm * 16 + r) * K + 2 * half;
    // B fragment source (B = W^T): W[n0 + r][k + 2*half + {0,1}]
    const float* wrow0 = W + (size_t)(tn4 * 64 + 0 * 16 + r) * K + 2 * half;
    const float* wrow1 = W + (size_t)(tn4 * 64 + 1 * 16 + r) * K + 2 * half;
    const float* wrow2 = W + (size_t)(tn4 * 64 + 2 * 16 + r) * K + 2 * half;
    const float* wrow3 = W + (size_t)(tn4 * 64 + 3 * 16 + r) * K + 2 * half;

    v8f acc0 = {}, acc1 = {}, acc2 = {}, acc3 = {};
    if (R) {
        // C/D layout: VGPR j -> row tm*16 + j + 8*half, col n0 + r
        const float* rb = R + (size_t)(tm * 16 + half * 8) * N + tn4 * 64 + r;
#pragma unroll
        for (int j = 0; j < 8; ++j) {
            acc0[j] = rb[(size_t)j * N + 0];
            acc1[j] = rb[(size_t)j * N + 16];
            acc2[j] = rb[(size_t)j * N + 32];
            acc3[j] = rb[(size_t)j * N + 48];
        }
    }

    for (int k = 0; k < K; k += 4) {
        v2f a  = *(const v2f*)(arow + k);
        v2f b0 = *(const v2f*)(wrow0 + k);
        v2f b1 = *(const v2f*)(wrow1 + k);
        v2f b2 = *(const v2f*)(wrow2 + k);
        v2f b3 = *(const v2f*)(wrow3 + k);
        acc0 = __builtin_amdgcn_wmma_f32_16x16x4_f32(false, a, false, b0, (short)0, acc0, false, false);
        acc1 = __builtin_amdgcn_wmma_f32_16x16x4_f32(false, a, false, b1, (short)0, acc1, false, false);
        acc2 = __builtin_amdgcn_wmma_f32_16x16x4_f32(false, a, false, b2, (short)0, acc2, false, false);
        acc3 = __builtin_amdgcn_wmma_f32_16x16x4_f32(false, a, false, b3, (short)0, acc3, false, false);
    }

    float* ob = O + (size_t)(tm * 16 + half * 8) * N + tn4 * 64 + r;
#pragma unroll
    for (int j = 0; j < 8; ++j) {
        ob[(size_t)j * N + 0]  = acc0[j];
        ob[(size_t)j * N + 16] = acc1[j];
        ob[(size_t)j * N + 32] = acc2[j];
        ob[(size_t)j * N + 48] = acc3[j];
    }
}

// ---------------------------------------------------------------------------
// Depthwise causal conv (k=4, left zero pad) over xp = xz[..., 0:DI], then SiLU
// ---------------------------------------------------------------------------
__global__ void k_conv_silu(const float* __restrict__ xz, const float* __restrict__ cw,
                            const float* __restrict__ cb, float* __restrict__ xc) {
    int idx = blockIdx.x * blockDim.x + threadIdx.x;       // MROWS*DI threads
    int m = idx >> 9;
    int d = idx & 511;
    int t = m & (NSEQ - 1);
    float acc = cb[d];
    const float* wd = cw + (size_t)d * 4;
#pragma unroll
    for (int i = 0; i < 4; ++i) {
        int ti = t - 3 + i;
        if (ti >= 0) acc += xz[(size_t)(m - 3 + i) * (2 * DI) + d] * wd[i];
    }
    xc[(size_t)m * DI + d] = silu_f(acc);
}

// ---------------------------------------------------------------------------
// ssm[m,j] = sum_k xc[m,k] * Wx[j,k]   (N=33, K=512) — plain dot per output
// ---------------------------------------------------------------------------
__global__ void k_xproj(const float* __restrict__ xc, const float* __restrict__ wx,
                        float* __restrict__ ssm) {
    int idx = blockIdx.x * blockDim.x + threadIdx.x;
    if (idx >= MROWS * DS2) return;
    int m = idx / DS2;
    int j = idx % DS2;
    const float4* xr = (const float4*)(xc + (size_t)m * DI);
    const float4* wr = (const float4*)(wx + (size_t)j * DI);
    float acc = 0.0f;
#pragma unroll 4
    for (int k = 0; k < DI / 4; ++k) {
        float4 a = xr[k];
        float4 b = wr[k];
        acc += a.x * b.x + a.y * b.y + a.z * b.z + a.w * b.w;
    }
    ssm[(size_t)m * DS2 + j] = acc;
}

// ---------------------------------------------------------------------------
// Selective scan: one thread per (batch, channel, state); 16-lane shfl
// reduction for y = sum_s h*C. No barriers in the 4096-step loop.
// ---------------------------------------------------------------------------
__global__ void k_scan(const float* __restrict__ xc, const float* __restrict__ ssm,
                       const float* __restrict__ dtw, const float* __restrict__ dtb,
                       float* __restrict__ y) {
    int tid = blockIdx.x * blockDim.x + threadIdx.x;       // NBATCH*DI*NST = 65536
    int s = tid & 15;
    int d = (tid >> 4) & (DI - 1);
    int b = tid >> 13;
    float dw = dtw[d];
    float db = dtb[d];
    float As = -(float)(s + 1);
    float h = 0.0f;
    size_t base = (size_t)b * NSEQ;
    for (int t = 0; t < NSEQ; ++t) {
        size_t m = base + t;
        float xv = xc[m * DI + d];
        const float* sr = ssm + m * DS2;
        float Bv  = sr[s];
        float Cv  = sr[NST + s];
        float dtr = sr[2 * NST];
        float zz = dtr * dw + db;
        float dt = (zz > 20.0f) ? zz : log1pf(__expf(zz));  // softplus
        h = __expf(dt * As) * h + dt * xv * Bv;
        float p = h * Cv;
        p += __shfl_xor(p, 1);
        p += __shfl_xor(p, 2);
        p += __shfl_xor(p, 4);
        p += __shfl_xor(p, 8);
        if (s == 0) y[m * DI + d] = p;
    }
}

// ---------------------------------------------------------------------------
// y[m,d] *= silu(z[m,d]),  z = xz[m, DI + d]
// ---------------------------------------------------------------------------
__global__ void k_gate(float* __restrict__ y, const float* __restrict__ xz) {
    int idx = blockIdx.x * blockDim.x + threadIdx.x;       // MROWS*DI
    int m = idx >> 9;
    int d = idx & 511;
    y[(size_t)m * DI + d] *= silu_f(xz[(size_t)m * (2 * DI) + DI + d]);
}

// ---------------------------------------------------------------------------
// Final layernorm + output projection (OUT=2), wave per row
// ---------------------------------------------------------------------------
__global__ void k_final(const float* __restrict__ x, const float* __restrict__ g,
                        const float* __restrict__ b, const float* __restrict__ opw,
                        const float* __restrict__ opb, float* __restrict__ out, int M) {
    int wave = (blockIdx.x * blockDim.x + threadIdx.x) >> 5;
    int lane = threadIdx.x & 31;
    if (wave >= M) return;
    const float* row = x + (size_t)wave * DM;
    float v[8];
    float s = 0.0f;
#pragma unroll
    for (int i = 0; i < 8; ++i) { v[i] = row[i * 32 + lane]; s += v[i]; }
    s = wave_sum32(s);
    float mu = s * (1.0f / DM);
    float q = 0.0f;
#pragma unroll
    for (int i = 0; i < 8; ++i) { float d = v[i] - mu; q += d * d; }
    q = wave_sum32(q);
    float rs = rsqrtf(q * (1.0f / DM) + LEPS);
    float a0 = 0.0f, a1 = 0.0f;
#pragma unroll
    for (int i = 0; i < 8; ++i) {
        int c = i * 32 + lane;
        float xn = (v[i] - mu) * rs * g[c] + b[c];
        a0 += xn * opw[c];
        a1 += xn * opw[DM + c];
    }
    a0 = wave_sum32(a0);
    a1 = wave_sum32(a1);
    if (lane == 0) {
        out[(size_t)wave * 2 + 0] = a0 + opb[0];
        out[(size_t)wave * 2 + 1] = a1 + opb[1];
    }
}

// ---------------------------------------------------------------------------
extern "C" void kernel_launch(void* const* d_in, const int* in_sizes, int n_in,
                              void* d_out, int out_size, void* d_ws, size_t ws_size,
                              hipStream_t stream) {
    const float* z_seq   = (const float*)d_in[0];
    const float* ip_w    = (const float*)d_in[1];
    const float* ip_b    = (const float*)d_in[2];
    const float* norm_g  = (const float*)d_in[3];
    const float* norm_b  = (const float*)d_in[4];
    const float* inproj  = (const float*)d_in[5];
    const float* conv_w  = (const float*)d_in[6];
    const float* conv_b  = (const float*)d_in[7];
    const float* xproj   = (const float*)d_in[8];
    const float* dt_w    = (const float*)d_in[9];
    const float* dt_b    = (const float*)d_in[10];
    const float* outproj = (const float*)d_in[11];
    const float* onorm_g = (const float*)d_in[12];
    const float* onorm_b = (const float*)d_in[13];
    const float* op_w    = (const float*)d_in[14];
    const float* op_b    = (const float*)d_in[15];
    float* out = (float*)d_out;

    float* ws  = (float*)d_ws;
    float* x   = ws;                                 // MROWS*DM
    float* xn  = x  + (size_t)MROWS * DM;            // MROWS*DM
    float* xz  = xn + (size_t)MROWS * DM;            // MROWS*2*DI
    float* xc  = xz + (size_t)MROWS * 2 * DI;        // MROWS*DI
    float* yb  = xc + (size_t)MROWS * DI;            // MROWS*DI
    float* ssm = yb + (size_t)MROWS * DI;            // MROWS*DS2

    // x = z_seq @ ip_w.T + ip_b
    k_input_proj<<<(MROWS * DM) / 256, 256, 0, stream>>>(z_seq, ip_w, ip_b, x);

    for (int l = 0; l < 2; ++l) {
        k_layernorm<<<MROWS / 8, 256, 0, stream>>>(x, norm_g + l * DM, norm_b + l * DM, xn, MROWS);

        // xz = xn @ Wi^T  : M=32768, N=1024, K=256
        {
            int waves = (MROWS / 16) * ((2 * DI) / 64);
            k_wmma_gemm_nt<<<waves / 8, 256, 0, stream>>>(
                xn, inproj + (size_t)l * 2 * DI * DM, nullptr, xz, MROWS, 2 * DI, DM);
        }

        k_conv_silu<<<(MROWS * DI) / 256, 256, 0, stream>>>(
            xz, conv_w + (size_t)l * DI * 4, conv_b + (size_t)l * DI, xc);

        k_xproj<<<(MROWS * DS2 + 255) / 256, 256, 0, stream>>>(
            xc, xproj + (size_t)l * DS2 * DI, ssm);

        k_scan<<<(NBATCH * DI * NST) / 256, 256, 0, stream>>>(
            xc, ssm, dt_w + (size_t)l * DI, dt_b + (size_t)l * DI, yb);

        k_gate<<<(MROWS * DI) / 256, 256, 0, stream>>>(yb, xz);

        // x = yb @ Wo^T + x  : M=32768, N=256, K=512 (residual as WMMA C-init)
        {
            int waves = (MROWS / 16) * (DM / 64);
            k_wmma_gemm_nt<<<waves / 8, 256, 0, stream>>>(
                yb, outproj + (size_t)l * DM * DI, x, x, MROWS, DM, DI);
        }
    }

    k_final<<<MROWS / 8, 256, 0, stream>>>(x, onorm_g, onorm_b, op_w, op_b, out, MROWS);
}